// GATMergeQuestionG_38482906972460
// MI455X (gfx1250) — compile-verified
//
#include <hip/hip_runtime.h>

// ---------------- problem sizes ----------------
#define Bn 32
#define Ln 32
#define Dn 300
#define Hn 512
#define Pn 512
#define NHEAD 4
#define HDn 128
#define NIn 128
#define NEn 512
#define NKn 128
#define NKEn 512
#define TLn 8
#define An 2000
#define NNODES 256   // NI + NK
#define NEDGES 1024  // NE + NKE
#define G4H (4*Hn)   // 2048

typedef __attribute__((ext_vector_type(2))) float v2f;
typedef __attribute__((ext_vector_type(8))) float v8f;

// =====================================================================
// Generic fp32 WMMA GEMM:  C[M,N] = act( A[M,K] * B[K,N] (+ addsrc) (+ bias) )
// block = 256 threads = 8 waves; block tile = 16(M) x 128(N); K staged 32 deep in LDS.
// Register-staged double buffering: next chunk's global loads are issued before the
// current chunk's 8 V_WMMA_F32_16X16X4_F32 ops, hiding HBM/L2 latency behind the
// matrix pipe. Interior chunks use float2/float4 (b64/b128) loads; edges fall back
// to guarded scalar loads with zero fill. __builtin_prefetch -> global_prefetch_b8.
// Fragment layout (ISA 7.12.2, 32-bit 16x4): lane l -> idx=l&15, Kpair=(l>>4)*2.
// =====================================================================
__global__ __launch_bounds__(256)
void wmma_gemm_kernel(const float* __restrict__ A, int lda,
                      const float* __restrict__ B, int ldb,
                      float* __restrict__ C, int ldc,
                      const float* __restrict__ addsrc, int ld_add,
                      const float* __restrict__ bias,
                      int M, int N, int K, int act)
{
    __shared__ float As[16][36];    // 16 x 32 (+pad)
    __shared__ float Bs[32][132];   // 32 x 128 (+pad)

    const int tid  = threadIdx.x;
    const int lane = tid & 31;
    const int wave = tid >> 5;
    const int m0   = blockIdx.y * 16;
    const int nblk = blockIdx.x * 128;
    const int mrow = lane & 15;     // row-in-tile for A frag / col for B frag
    const int kh   = lane >> 4;     // K half selector
    const int ncol = nblk + wave * 16 + mrow;   // this lane's output column

    // staging assignment: A -> 16x32 tile, thread owns one float2
    const int ar = tid >> 4;            // 0..15
    const int ac = (tid & 15) * 2;      // 0..30
    // B -> 32x128 tile, thread owns 4 float4s: f = tid + i*256, row=f>>5, col4=(f&31)*4

    v8f acc;
#pragma unroll
    for (int j = 0; j < 8; ++j) acc[j] = 0.f;
    if (addsrc) {
#pragma unroll
        for (int j = 0; j < 8; ++j) {
            int m = m0 + kh * 8 + j;
            if (ncol < N && m < M) acc[j] = addsrc[(size_t)m * ld_add + ncol];
        }
    }

    float  a0 = 0.f, a1 = 0.f;
    float4 bq0, bq1, bq2, bq3;
    bq0 = bq1 = bq2 = bq3 = make_float4(0.f, 0.f, 0.f, 0.f);

    auto load_chunk = [&](int k0) {
        const int gm = m0 + ar, gk = k0 + ac;
        const bool full = (k0 + 32 <= K) && (nblk + 128 <= N) && (m0 + 16 <= M);
        if (full) {
            // interior: vector loads (b64 for A frag, b128 x4 for B frag)
            const float2 av = *(const float2*)(A + (size_t)gm * lda + gk);
            a0 = av.x; a1 = av.y;
            {
                int f, r, c4;
                f = tid;           r = f >> 5; c4 = (f & 31) * 4;
                bq0 = *(const float4*)(B + (size_t)(k0 + r) * ldb + nblk + c4);
                f = tid + 256;     r = f >> 5; c4 = (f & 31) * 4;
                bq1 = *(const float4*)(B + (size_t)(k0 + r) * ldb + nblk + c4);
                f = tid + 512;     r = f >> 5; c4 = (f & 31) * 4;
                bq2 = *(const float4*)(B + (size_t)(k0 + r) * ldb + nblk + c4);
                f = tid + 768;     r = f >> 5; c4 = (f & 31) * 4;
                bq3 = *(const float4*)(B + (size_t)(k0 + r) * ldb + nblk + c4);
            }
        } else {
            // edge: guarded scalar loads, zero fill (zeros contribute nothing)
            a0 = (gm < M && gk < K)     ? A[(size_t)gm * lda + gk]     : 0.f;
            a1 = (gm < M && gk + 1 < K) ? A[(size_t)gm * lda + gk + 1] : 0.f;
            float4* dstq[4] = { &bq0, &bq1, &bq2, &bq3 };
#pragma unroll
            for (int i = 0; i < 4; ++i) {
                int f = tid + i * 256;
                int r = f >> 5, c4 = (f & 31) * 4;
                int gkr = k0 + r;
                const float* bp = B + (size_t)gkr * ldb + nblk + c4;
                float4 v;
                v.x = (gkr < K && nblk + c4 + 0 < N) ? bp[0] : 0.f;
                v.y = (gkr < K && nblk + c4 + 1 < N) ? bp[1] : 0.f;
                v.z = (gkr < K && nblk + c4 + 2 < N) ? bp[2] : 0.f;
                v.w = (gkr < K && nblk + c4 + 3 < N) ? bp[3] : 0.f;
                *dstq[i] = v;
            }
        }
    };
    auto store_chunk = [&]() {
        As[ar][ac] = a0; As[ar][ac + 1] = a1;
        float4 vq[4] = { bq0, bq1, bq2, bq3 };
#pragma unroll
        for (int i = 0; i < 4; ++i) {
            int f = tid + i * 256;
            int r = f >> 5, c4 = (f & 31) * 4;
            Bs[r][c4 + 0] = vq[i].x; Bs[r][c4 + 1] = vq[i].y;
            Bs[r][c4 + 2] = vq[i].z; Bs[r][c4 + 3] = vq[i].w;
        }
    };

    load_chunk(0);
    store_chunk();
    __syncthreads();

    for (int k0 = 0; k0 < K; k0 += 32) {
        const int kn = k0 + 32;
        if (kn < K) {
            load_chunk(kn);                       // overlaps with WMMA below
            if (kn + 32 < K) {                    // prefetch chunk-after-next
                __builtin_prefetch(A + (size_t)(m0 + ar) * lda + (kn + 32 + ac), 0, 3);
                __builtin_prefetch(B + (size_t)(kn + 32 + (tid >> 5)) * ldb + nblk + (tid & 31) * 4, 0, 3);
            }
        }
#pragma unroll
        for (int kk = 0; kk < 8; ++kk) {
            v2f a, b;
            a.x = As[mrow][kk * 4 + kh * 2 + 0];
            a.y = As[mrow][kk * 4 + kh * 2 + 1];
            b.x = Bs[kk * 4 + kh * 2 + 0][wave * 16 + mrow];
            b.y = Bs[kk * 4 + kh * 2 + 1][wave * 16 + mrow];
            acc = __builtin_amdgcn_wmma_f32_16x16x4_f32(false, a, false, b,
                                                        (short)0, acc, false, false);
        }
        __syncthreads();
        if (kn < K) store_chunk();
        __syncthreads();
    }

    if (ncol < N) {
        float bv = bias ? bias[ncol] : 0.f;
#pragma unroll
        for (int j = 0; j < 8; ++j) {
            int m = m0 + kh * 8 + j;
            if (m < M) {
                float x = acc[j] + bv;
                if (act == 1) x = fmaxf(x, 0.f);
                C[(size_t)m * ldc + ncol] = x;
            }
        }
    }
}

// =====================================================================
// elementwise / gather kernels
// =====================================================================
__global__ void fill_kernel(float* p, size_t n, float v)
{
    size_t i = (size_t)blockIdx.x * blockDim.x + threadIdx.x;
    size_t st = (size_t)gridDim.x * blockDim.x;
    for (; i < n; i += st) p[i] = v;
}

__global__ void embed_q_kernel(const float* __restrict__ we, const int* __restrict__ q,
                               float* __restrict__ X)
{
    int idx = blockIdx.x * blockDim.x + threadIdx.x;
    if (idx >= Bn * Ln * Dn) return;
    int d = idx % Dn, bl = idx / Dn;
    X[idx] = we[(size_t)q[bl] * Dn + d];
}

// mean over TL tokens + mask flag (sum(tokens)==TL)
__global__ void embed_mean_kernel(const float* __restrict__ we, const int* __restrict__ toks,
                                  float* __restrict__ out, int* __restrict__ maskout, int rows)
{
    int idx = blockIdx.x * blockDim.x + threadIdx.x;
    if (idx >= rows * Dn) return;
    int d = idx % Dn, r = idx / Dn;
    const int* t = toks + (size_t)r * TLn;
    float s = 0.f; int ssum = 0;
#pragma unroll
    for (int j = 0; j < TLn; ++j) { int tk = t[j]; s += we[(size_t)tk * Dn + d]; ssum += tk; }
    out[idx] = s * (1.f / TLn);
    if (d == 0) maskout[r] = (ssum == TLn) ? 1 : 0;
}

__global__ void lstm_gate_kernel(const float* __restrict__ g, const float* __restrict__ bvec,
                                 const int* __restrict__ qmask, int t,
                                 float* __restrict__ h, float* __restrict__ c)
{
    int idx = blockIdx.x * blockDim.x + threadIdx.x;
    if (idx >= Bn * Hn) return;
    int b = idx / Hn, u = idx % Hn;
    const float* gr = g + (size_t)b * G4H;
    float i_ = gr[u]          + bvec[u];
    float f_ = gr[Hn + u]     + bvec[Hn + u];
    float g_ = gr[2 * Hn + u] + bvec[2 * Hn + u];
    float o_ = gr[3 * Hn + u] + bvec[3 * Hn + u];
    float si = 1.f / (1.f + expf(-i_));
    float sf = 1.f / (1.f + expf(-f_));
    float so = 1.f / (1.f + expf(-o_));
    float c2 = sf * c[idx] + si * tanhf(g_);
    float h2 = so * tanhf(c2);
    float m = (float)qmask[b * Ln + t];
    h[idx] = m * h2 + (1.f - m) * h[idx];
    c[idx] = m * c2 + (1.f - m) * c[idx];
}

// one wave per row: v = sum_p tanh(qW[b,p]+F[row,p]+bf[p])*Wv[p] + bv; masked fill
__global__ __launch_bounds__(256)
void att_reduce_kernel(const float* __restrict__ F, const float* __restrict__ qW,
                       const float* __restrict__ bf, const float* __restrict__ Wv,
                       const float* __restrict__ bvp, const int* __restrict__ mask,
                       float fill, float* __restrict__ out, int out_stride_b, int out_off,
                       int rows_per_b, int rows_total)
{
    int wid = (blockIdx.x * blockDim.x + threadIdx.x) >> 5;
    int lane = threadIdx.x & 31;
    if (wid >= rows_total) return;
    int b = wid / rows_per_b, i = wid - b * rows_per_b;
    const float* Fr = F + (size_t)wid * Pn;
    const float* qb = qW + (size_t)b * Pn;
    float s = 0.f;
    for (int p = lane; p < Pn; p += 32) s += tanhf(qb[p] + Fr[p] + bf[p]) * Wv[p];
    for (int o = 16; o; o >>= 1) s += __shfl_xor(s, o, 32);
    if (lane == 0) {
        float v = s + bvp[0];
        out[(size_t)b * out_stride_b + out_off + i] = mask[wid] ? fill : v;
    }
}

__global__ __launch_bounds__(256)
void softmax_rows_kernel(float* __restrict__ x, int n)
{
    __shared__ float red[256];
    float* row = x + (size_t)blockIdx.x * n;
    int tid = threadIdx.x;
    float mx = -3.0e38f;
    for (int i = tid; i < n; i += 256) mx = fmaxf(mx, row[i]);
    red[tid] = mx; __syncthreads();
    for (int s = 128; s; s >>= 1) { if (tid < s) red[tid] = fmaxf(red[tid], red[tid + s]); __syncthreads(); }
    mx = red[0]; __syncthreads();
    float sum = 0.f;
    for (int i = tid; i < n; i += 256) { float e = expf(row[i] - mx); row[i] = e; sum += e; }
    red[tid] = sum; __syncthreads();
    for (int s = 128; s; s >>= 1) { if (tid < s) red[tid] += red[tid + s]; __syncthreads(); }
    float inv = 1.f / red[0];
    for (int i = tid; i < n; i += 256) row[i] *= inv;
}

__global__ __launch_bounds__(256)
void rownorm_kernel(const float* __restrict__ x, float* __restrict__ out, int rows, int dim)
{
    int wid = (blockIdx.x * blockDim.x + threadIdx.x) >> 5;
    int lane = threadIdx.x & 31;
    if (wid >= rows) return;
    const float* r = x + (size_t)wid * dim;
    float s = 0.f;
    for (int d = lane; d < dim; d += 32) { float v = r[d]; s += v * v; }
    for (int o = 16; o; o >>= 1) s += __shfl_xor(s, o, 32);
    if (lane == 0) out[wid] = sqrtf(s);
}

// argmax over (inf_i . knf_j)/((|inf_i|+eps)(|knf_j|+eps)); tie -> lowest flat index
__global__ __launch_bounds__(256)
void sim_argmax_kernel(const float* __restrict__ inf, const float* __restrict__ knf,
                       const float* __restrict__ na, const float* __restrict__ nb,
                       int* __restrict__ img_id, int* __restrict__ kg_id)
{
    int b = blockIdx.x, tid = threadIdx.x;
    const float* A  = inf + (size_t)b * NIn * Dn;
    const float* Bm = knf + (size_t)b * NKn * Dn;
    const float* ra = na + b * NIn;
    const float* rb = nb + b * NKn;
    float best = -3.0e38f; int bidx = 0;
    for (int p = tid; p < NIn * NKn; p += 256) {
        int i = p / NKn, j = p - i * NKn;
        const float* ar = A + (size_t)i * Dn;
        const float* br = Bm + (size_t)j * Dn;
        float s = 0.f;
        for (int d = 0; d < Dn; ++d) s += ar[d] * br[d];
        s /= ((ra[i] + 1e-8f) * (rb[j] + 1e-8f));
        if (s > best) { best = s; bidx = p; }
    }
    __shared__ float bv[256]; __shared__ int bi[256];
    bv[tid] = best; bi[tid] = bidx; __syncthreads();
    for (int s = 128; s; s >>= 1) {
        if (tid < s) {
            if (bv[tid + s] > bv[tid] || (bv[tid + s] == bv[tid] && bi[tid + s] < bi[tid])) {
                bv[tid] = bv[tid + s]; bi[tid] = bi[tid + s];
            }
        }
        __syncthreads();
    }
    if (tid == 0) { img_id[b] = bi[0] / NKn; kg_id[b] = bi[0] % NKn; }
}

__global__ void build_edges_kernel(const int* __restrict__ i1, const int* __restrict__ i2,
                                   const int* __restrict__ k1, const int* __restrict__ k2,
                                   const int* __restrict__ img_id, const int* __restrict__ kg_id,
                                   int* __restrict__ src, int* __restrict__ dst)
{
    int idx = blockIdx.x * blockDim.x + threadIdx.x;
    if (idx >= Bn * NEDGES) return;
    int b = idx / NEDGES, e = idx - b * NEDGES;
    int s, d;
    if (e < NEn) { s = i1[b * NEn + e]; d = i2[b * NEn + e]; }
    else {
        int ee = e - NEn;
        s = k1[b * NKEn + ee] + NIn;
        d = k2[b * NKEn + ee] + NIn;
        int kid = kg_id[b] + NIn, iid = img_id[b];
        if (s == kid) s = iid;
        if (d == kid) d = iid;
    }
    src[idx] = s; dst[idx] = d;
}

__global__ void scale_nodes_kernel(const float* __restrict__ inf, const float* __restrict__ knf,
                                   const float* __restrict__ natt, float* __restrict__ out)
{
    int idx = blockIdx.x * blockDim.x + threadIdx.x;
    if (idx >= Bn * NNODES * Dn) return;
    int d = idx % Dn, br = idx / Dn;
    int b = br / NNODES, r = br - b * NNODES;
    float v = (r < NIn) ? inf[((size_t)b * NIn + r) * Dn + d]
                        : knf[((size_t)b * NKn + (r - NIn)) * Dn + d];
    out[idx] = v * natt[b * NNODES + r];
}

__global__ void scale_edges_kernel(const float* __restrict__ ief, const float* __restrict__ kef,
                                   const float* __restrict__ eatt, float* __restrict__ out)
{
    int idx = blockIdx.x * blockDim.x + threadIdx.x;
    if (idx >= Bn * NEDGES * Dn) return;
    int d = idx % Dn, br = idx / Dn;
    int b = br / NEDGES, r = br - b * NEDGES;
    float v = (r < NEn) ? ief[((size_t)b * NEn + r) * Dn + d]
                        : kef[((size_t)b * NKEn + (r - NEn)) * Dn + d];
    out[idx] = v * eatt[b * NEDGES + r];
}

// per-(row,head): d0 = z_row . a[h,off0:off0+HD]; d1 likewise (optional)
__global__ void gat_dots_kernel(const float* __restrict__ z, const float* __restrict__ a,
                                int off0, int off1, float* __restrict__ d0,
                                float* __restrict__ d1, int rows)
{
    int idx = blockIdx.x * blockDim.x + threadIdx.x;
    if (idx >= rows * NHEAD) return;
    int r = idx / NHEAD, h = idx - r * NHEAD;
    const float* zr = z + (size_t)r * (NHEAD * HDn) + h * HDn;
    const float* a0 = a + h * (3 * HDn) + off0;
    const float* a1 = a + h * (3 * HDn) + off1;
    float s0 = 0.f, s1 = 0.f;
    for (int d = 0; d < HDn; ++d) { float zz = zr[d]; s0 += zz * a0[d]; s1 += zz * a1[d]; }
    d0[idx] = s0;
    if (d1) d1[idx] = s1;
}

__global__ void gat_score_kernel(const float* __restrict__ ds, const float* __restrict__ dd,
                                 const float* __restrict__ de, const int* __restrict__ src,
                                 const int* __restrict__ dst, float* __restrict__ score)
{
    int idx = blockIdx.x * blockDim.x + threadIdx.x;
    if (idx >= Bn * NEDGES * NHEAD) return;
    int h = idx & 3;
    int be = idx >> 2;              // b*NEDGES + e
    int b = be / NEDGES;
    int s = src[be], d = dst[be];
    float x = ds[((size_t)b * NNODES + s) * NHEAD + h]
            + dd[((size_t)b * NNODES + d) * NHEAD + h]
            + de[(size_t)be * NHEAD + h];
    score[idx] = x > 0.f ? x : 0.2f * x;
}

// one wave per (b,node): ballot-compact incoming-edge list, segment softmax, gather z[src]
__global__ __launch_bounds__(256)
void gat_aggregate_kernel(const float* __restrict__ z, const float* __restrict__ score,
                          const int* __restrict__ src, const int* __restrict__ dst,
                          float* __restrict__ hout)
{
    __shared__ int lists[8][NEDGES];
    int wave = threadIdx.x >> 5, lane = threadIdx.x & 31;
    int gw = blockIdx.x * 8 + wave;           // grid is exact: Bn*NNODES/8 blocks
    int b = gw >> 8, n = gw & 255;
    const int* dstb = dst + b * NEDGES;
    const int* srcb = src + b * NEDGES;
    const float* scb = score + (size_t)b * NEDGES * NHEAD;

    int cnt = 0;
    for (int e0 = 0; e0 < NEDGES; e0 += 32) {
        int e = e0 + lane;
        bool m = (dstb[e] == n);
        unsigned mask32 = (unsigned)__ballot(m);
        int pre = __popc(mask32 & ((1u << lane) - 1u));
        if (m) lists[wave][cnt + pre] = e;
        cnt += __popc(mask32);
    }

    float mx[NHEAD];
#pragma unroll
    for (int h = 0; h < NHEAD; ++h) mx[h] = -3.0e38f;
    for (int i = lane; i < cnt; i += 32) {
        int e = lists[wave][i];
#pragma unroll
        for (int h = 0; h < NHEAD; ++h) mx[h] = fmaxf(mx[h], scb[(size_t)e * NHEAD + h]);
    }
#pragma unroll
    for (int h = 0; h < NHEAD; ++h)
        for (int o = 16; o; o >>= 1) mx[h] = fmaxf(mx[h], __shfl_xor(mx[h], o, 32));

    float den[NHEAD];
#pragma unroll
    for (int h = 0; h < NHEAD; ++h) den[h] = 0.f;
    for (int i = lane; i < cnt; i += 32) {
        int e = lists[wave][i];
#pragma unroll
        for (int h = 0; h < NHEAD; ++h) den[h] += expf(scb[(size_t)e * NHEAD + h] - mx[h]);
    }
#pragma unroll
    for (int h = 0; h < NHEAD; ++h)
        for (int o = 16; o; o >>= 1) den[h] += __shfl_xor(den[h], o, 32);

    float acc[16];
#pragma unroll
    for (int j = 0; j < 16; ++j) acc[j] = 0.f;
    for (int i = 0; i < cnt; ++i) {
        int e = lists[wave][i];
        int s = srcb[e];
        float coef[NHEAD];
#pragma unroll
        for (int h = 0; h < NHEAD; ++h)
            coef[h] = expf(scb[(size_t)e * NHEAD + h] - mx[h]) / (den[h] + 1e-9f);
        const float* zs = z + ((size_t)b * NNODES + s) * (NHEAD * HDn);
#pragma unroll
        for (int j = 0; j < 16; ++j) {
            int d = lane + 32 * j;
            acc[j] += coef[d >> 7] * zs[d];
        }
    }
    float* ho = hout + ((size_t)b * NNODES + n) * (NHEAD * HDn);
#pragma unroll
    for (int j = 0; j < 16; ++j) {
        int d = lane + 32 * j;
        float x = acc[j];
        ho[d] = x > 0.f ? x : (expf(x) - 1.f);
    }
}

__global__ void elu_inplace_kernel(float* x, size_t n)
{
    size_t i = (size_t)blockIdx.x * blockDim.x + threadIdx.x;
    size_t st = (size_t)gridDim.x * blockDim.x;
    for (; i < n; i += st) { float v = x[i]; x[i] = v > 0.f ? v : (expf(v) - 1.f); }
}

__global__ void colsum_kernel(const float* __restrict__ x, float* __restrict__ out,
                              int rowsPerB, int dim)
{
    int idx = blockIdx.x * blockDim.x + threadIdx.x;
    if (idx >= Bn * dim) return;
    int b = idx / dim, d = idx - b * dim;
    const float* p = x + ((size_t)b * rowsPerB) * dim + d;
    float s = 0.f;
    for (int r = 0; r < rowsPerB; ++r) s += p[(size_t)r * dim];
    out[idx] = s;
}

__global__ void combine_kernel(const float* __restrict__ nsum, const float* __restrict__ qn,
                               const float* __restrict__ efp, const float* __restrict__ qe,
                               float* __restrict__ cat)
{
    int idx = blockIdx.x * blockDim.x + threadIdx.x;
    if (idx >= Bn * 2 * Hn) return;
    int b = idx / (2 * Hn), j = idx - b * 2 * Hn;
    cat[idx] = (j < Hn) ? nsum[b * Hn + j] * qn[b * Hn + j]
                        : efp[b * Hn + (j - Hn)] * qe[b * Hn + (j - Hn)];
}

// =====================================================================
// host orchestration
// =====================================================================
static inline void launch_gemm(hipStream_t s, const float* A, int lda, const float* B, int ldb,
                               float* C, int ldc, const float* addsrc, int ld_add,
                               const float* bias, int M, int N, int K, int act)
{
    dim3 grid((N + 127) / 128, (M + 15) / 16);
    wmma_gemm_kernel<<<grid, 256, 0, s>>>(A, lda, B, ldb, C, ldc, addsrc, ld_add, bias, M, N, K, act);
}

extern "C" void kernel_launch(void* const* d_in, const int* in_sizes, int n_in,
                              void* d_out, int out_size, void* d_ws, size_t ws_size,
                              hipStream_t stream)
{
    (void)in_sizes; (void)n_in; (void)out_size; (void)ws_size;
    // ---- params (recursive dict-insertion order) ----
    const float* word_emb = (const float*)d_in[0];
    const float* Wih  = (const float*)d_in[1];
    const float* Whh  = (const float*)d_in[2];
    const float* lstmb= (const float*)d_in[3];
    // att sets: img_node, img_edge, kg_node, kg_edge ; each Wq,bq,Wf,bf,Wv,bv
    const float* g1Wn = (const float*)d_in[28];
    const float* g1We = (const float*)d_in[29];
    const float* g1a  = (const float*)d_in[30];
    const float* g2Wn = (const float*)d_in[31];
    const float* g2We = (const float*)d_in[32];
    const float* g2a  = (const float*)d_in[33];
    const float* qnW  = (const float*)d_in[34];
    const float* qnb  = (const float*)d_in[35];
    const float* qeW  = (const float*)d_in[36];
    const float* qeb  = (const float*)d_in[37];
    const float* epW  = (const float*)d_in[38];
    const float* epb  = (const float*)d_in[39];
    const float* midW = (const float*)d_in[40];
    const float* midb = (const float*)d_in[41];
    const float* preW = (const float*)d_in[42];
    const float* preb = (const float*)d_in[43];
    const int* question = (const int*)d_in[44];
    const int* qmask    = (const int*)d_in[45];
    const int* img_nodes= (const int*)d_in[46];
    const int* img_edges= (const int*)d_in[47];
    const int* kg_nodes = (const int*)d_in[48];
    const int* kg_edges = (const int*)d_in[49];
    const int* i1 = (const int*)d_in[50];
    const int* i2 = (const int*)d_in[51];
    const int* k1 = (const int*)d_in[52];
    const int* k2 = (const int*)d_in[53];
    float* out = (float*)d_out;

    // ---- workspace layout (floats) ----
    float* wsf = (float*)d_ws;
    size_t wo = 0;
    auto ALLOC = [&](size_t n) -> float* { float* p = wsf + wo; wo += n; return p; };

    float* NODES = ALLOC((size_t)Bn * NNODES * Dn);   // also aliases X (question embeds)
    float* EDGES = ALLOC((size_t)Bn * NEDGES * Dn);   // also aliases FATT
    float* X    = NODES;                              // Bn*Ln*Dn <= NODES size
    float* FATT = EDGES;                              // Bn*NEn*Pn <= EDGES size
    float* XG   = ALLOC((size_t)Bn * Ln * G4H);
    float* Hh   = ALLOC((size_t)Bn * Hn);
    float* Cc   = ALLOC((size_t)Bn * Hn);
    float* Gg   = ALLOC((size_t)Bn * G4H);
    float* INF  = ALLOC((size_t)Bn * NIn * Dn);
    float* KNF  = ALLOC((size_t)Bn * NKn * Dn);
    float* IEF  = ALLOC((size_t)Bn * NEn * Dn);
    float* KEF  = ALLOC((size_t)Bn * NKEn * Dn);
    int*   MASKI  = (int*)ALLOC(Bn * NIn);
    int*   MASKIE = (int*)ALLOC(Bn * NEn);
    int*   MASKK  = (int*)ALLOC(Bn * NKn);
    int*   MASKKE = (int*)ALLOC(Bn * NKEn);
    float* QW   = ALLOC((size_t)Bn * Pn);
    float* NATT = ALLOC((size_t)Bn * NNODES);
    float* EATT = ALLOC((size_t)Bn * NEDGES);
    float* NORMI= ALLOC(Bn * NIn);
    float* NORMK= ALLOC(Bn * NKn);
    int*   IMGID= (int*)ALLOC(Bn);
    int*   KGID = (int*)ALLOC(Bn);
    int*   SRC  = (int*)ALLOC(Bn * NEDGES);
    int*   DST  = (int*)ALLOC(Bn * NEDGES);
    float* Z    = ALLOC((size_t)Bn * NNODES * 512);
    float* ZE   = ALLOC((size_t)Bn * NEDGES * 512);
    float* H1   = ALLOC((size_t)Bn * NNODES * 512);
    float* ZE2  = ALLOC((size_t)Bn * NEDGES * 512);
    float* DS_  = ALLOC((size_t)Bn * NNODES * NHEAD);
    float* DD_  = ALLOC((size_t)Bn * NNODES * NHEAD);
    float* DE_  = ALLOC((size_t)Bn * NEDGES * NHEAD);
    float* SCORE= ALLOC((size_t)Bn * NEDGES * NHEAD);
    float* NSUM = ALLOC((size_t)Bn * Hn);
    float* ESUM = ALLOC((size_t)Bn * Hn);
    float* QN   = ALLOC((size_t)Bn * Hn);
    float* QE   = ALLOC((size_t)Bn * Hn);
    float* EFP  = ALLOC((size_t)Bn * Hn);
    float* CAT  = ALLOC((size_t)Bn * 2 * Hn);
    float* MID  = ALLOC((size_t)Bn * 3 * Hn);

    // ---- 1. question embeddings + X@Wih precompute ----
    embed_q_kernel<<<(Bn * Ln * Dn + 255) / 256, 256, 0, stream>>>(word_emb, question, X);
    launch_gemm(stream, X, Dn, Wih, G4H, XG, G4H, nullptr, 0, nullptr, Bn * Ln, G4H, Dn, 0);

    // ---- 2. graph feature embeddings (mean over TL) ----
    embed_mean_kernel<<<(Bn * NIn * Dn + 255) / 256, 256, 0, stream>>>(word_emb, img_nodes, INF, MASKI, Bn * NIn);
    embed_mean_kernel<<<(Bn * NEn * Dn + 255) / 256, 256, 0, stream>>>(word_emb, img_edges, IEF, MASKIE, Bn * NEn);
    embed_mean_kernel<<<(Bn * NKn * Dn + 255) / 256, 256, 0, stream>>>(word_emb, kg_nodes, KNF, MASKK, Bn * NKn);
    embed_mean_kernel<<<(Bn * NKEn * Dn + 255) / 256, 256, 0, stream>>>(word_emb, kg_edges, KEF, MASKKE, Bn * NKEn);

    // ---- 3. LSTM (32 steps), q = final h ----
    fill_kernel<<<64, 256, 0, stream>>>(Hh, (size_t)Bn * Hn, 0.f);
    fill_kernel<<<64, 256, 0, stream>>>(Cc, (size_t)Bn * Hn, 0.f);
    for (int t = 0; t < Ln; ++t) {
        launch_gemm(stream, Hh, Hn, Whh, G4H, Gg, G4H,
                    XG + (size_t)t * G4H, Ln * G4H, nullptr, Bn, G4H, Hn, 0);
        lstm_gate_kernel<<<(Bn * Hn + 255) / 256, 256, 0, stream>>>(Gg, lstmb, qmask, t, Hh, Cc);
    }

    // ---- 4. additive attention (4 sets) ----
    const float* feats[4] = { INF, IEF, KNF, KEF };
    const int    rowsA[4] = { NIn, NEn, NKn, NKEn };
    const int*   masks[4] = { MASKI, MASKIE, MASKK, MASKKE };
    float*       outs[4]  = { NATT, EATT, NATT, EATT };
    const int    ostr[4]  = { NNODES, NEDGES, NNODES, NEDGES };
    const int    ooff[4]  = { 0, 0, NIn, NEn };
    const float  fills[4] = { -1e9f, -1e9f, -1e9f, 1e-9f };
    for (int s = 0; s < 4; ++s) {
        const float* Wq = (const float*)d_in[4 + s * 6 + 0];
        const float* bq = (const float*)d_in[4 + s * 6 + 1];
        const float* Wf = (const float*)d_in[4 + s * 6 + 2];
        const float* bf = (const float*)d_in[4 + s * 6 + 3];
        const float* Wv = (const float*)d_in[4 + s * 6 + 4];
        const float* bv = (const float*)d_in[4 + s * 6 + 5];
        launch_gemm(stream, Hh, Hn, Wq, Pn, QW, Pn, nullptr, 0, bq, Bn, Pn, Hn, 0);
        int rows = Bn * rowsA[s];
        launch_gemm(stream, feats[s], Dn, Wf, Pn, FATT, Pn, nullptr, 0, nullptr, rows, Pn, Dn, 0);
        att_reduce_kernel<<<rows / 8, 256, 0, stream>>>(FATT, QW, bf, Wv, bv, masks[s],
                                                        fills[s], outs[s], ostr[s], ooff[s],
                                                        rowsA[s], rows);
    }
    softmax_rows_kernel<<<Bn, 256, 0, stream>>>(NATT, NNODES);
    softmax_rows_kernel<<<Bn, 256, 0, stream>>>(EATT, NEDGES);

    // ---- 5. per-batch cosine argmax + graph build ----
    rownorm_kernel<<<(Bn * NIn) / 8, 256, 0, stream>>>(INF, NORMI, Bn * NIn, Dn);
    rownorm_kernel<<<(Bn * NKn) / 8, 256, 0, stream>>>(KNF, NORMK, Bn * NKn, Dn);
    sim_argmax_kernel<<<Bn, 256, 0, stream>>>(INF, KNF, NORMI, NORMK, IMGID, KGID);
    build_edges_kernel<<<(Bn * NEDGES + 255) / 256, 256, 0, stream>>>(i1, i2, k1, k2, IMGID, KGID, SRC, DST);
    scale_nodes_kernel<<<((size_t)Bn * NNODES * Dn + 255) / 256, 256, 0, stream>>>(INF, KNF, NATT, NODES);
    scale_edges_kernel<<<((size_t)Bn * NEDGES * Dn + 255) / 256, 256, 0, stream>>>(IEF, KEF, EATT, EDGES);

    // ---- 6. GAT layers ----
    // layer 1 (Fin = D)
    launch_gemm(stream, NODES, Dn, g1Wn, 512, Z, 512, nullptr, 0, nullptr, Bn * NNODES, 512, Dn, 0);
    launch_gemm(stream, EDGES, Dn, g1We, 512, ZE, 512, nullptr, 0, nullptr, Bn * NEDGES, 512, Dn, 0);
    gat_dots_kernel<<<(Bn * NNODES * NHEAD + 255) / 256, 256, 0, stream>>>(Z, g1a, 0, HDn, DS_, DD_, Bn * NNODES);
    gat_dots_kernel<<<(Bn * NEDGES * NHEAD + 255) / 256, 256, 0, stream>>>(ZE, g1a, 2 * HDn, 2 * HDn, DE_, nullptr, Bn * NEDGES);
    gat_score_kernel<<<(Bn * NEDGES * NHEAD + 255) / 256, 256, 0, stream>>>(DS_, DD_, DE_, SRC, DST, SCORE);
    gat_aggregate_kernel<<<(Bn * NNODES) / 8, 256, 0, stream>>>(Z, SCORE, SRC, DST, H1);
    elu_inplace_kernel<<<4096, 256, 0, stream>>>(ZE, (size_t)Bn * NEDGES * 512);   // e1 in place
    // layer 2 (Fin = H)
    launch_gemm(stream, H1, 512, g2Wn, 512, Z, 512, nullptr, 0, nullptr, Bn * NNODES, 512, 512, 0);
    launch_gemm(stream, ZE, 512, g2We, 512, ZE2, 512, nullptr, 0, nullptr, Bn * NEDGES, 512, 512, 0);
    gat_dots_kernel<<<(Bn * NNODES * NHEAD + 255) / 256, 256, 0, stream>>>(Z, g2a, 0, HDn, DS_, DD_, Bn * NNODES);
    gat_dots_kernel<<<(Bn * NEDGES * NHEAD + 255) / 256, 256, 0, stream>>>(ZE2, g2a, 2 * HDn, 2 * HDn, DE_, nullptr, Bn * NEDGES);
    gat_score_kernel<<<(Bn * NEDGES * NHEAD + 255) / 256, 256, 0, stream>>>(DS_, DD_, DE_, SRC, DST, SCORE);
    gat_aggregate_kernel<<<(Bn * NNODES) / 8, 256, 0, stream>>>(Z, SCORE, SRC, DST, H1);   // h2 -> H1
    elu_inplace_kernel<<<4096, 256, 0, stream>>>(ZE2, (size_t)Bn * NEDGES * 512);          // e2 in place

    // ---- 7. readout head ----
    colsum_kernel<<<(Bn * Hn + 255) / 256, 256, 0, stream>>>(H1, NSUM, NNODES, Hn);
    colsum_kernel<<<(Bn * Hn + 255) / 256, 256, 0, stream>>>(ZE2, ESUM, NEDGES, Hn);
    launch_gemm(stream, Hh, Hn, qnW, Hn, QN, Hn, nullptr, 0, qnb, Bn, Hn, Hn, 0);
    launch_gemm(stream, Hh, Hn, qeW, Hn, QE, Hn, nullptr, 0, qeb, Bn, Hn, Hn, 0);
    launch_gemm(stream, ESUM, Hn, epW, Hn, EFP, Hn, nullptr, 0, epb, Bn, Hn, Hn, 0);
    combine_kernel<<<(Bn * 2 * Hn + 255) / 256, 256, 0, stream>>>(NSUM, QN, EFP, QE, CAT);
    launch_gemm(stream, CAT, 2 * Hn, midW, 3 * Hn, MID, 3 * Hn, nullptr, 0, midb, Bn, 3 * Hn, 2 * Hn, 1);
    launch_gemm(stream, MID, 3 * Hn, preW, An, out, An, nullptr, 0, preb, Bn, An, 3 * Hn, 0);
}